// SparseAdam_35167192220106
// MI455X (gfx1250) — compile-verified
//
#include <hip/hip_runtime.h>
#include <stdint.h>

// ---------------------------------------------------------------------------
// Fused sparse-Adam for an embedding table (V=1e6, D=64, N=524288).
// Phase 1: zero gsum/cnt accumulators (aliased onto emb_new / step_new output
//          regions -> no workspace needed).
// Phase 2: scatter grad rows with hardware FP32 global atomics (b128 loads,
//          16 lanes per row).
// Phase 3: per-row Adam finalize (16 lanes per row), streaming inputs through
//          LDS via CDNA5 async global->LDS b128 loads (ASYNCcnt).
// ---------------------------------------------------------------------------

#define D_DIM 64

#define ADAM_LR    0.01f
#define ADAM_B1    0.9f
#define ADAM_B2    0.999f
#define ADAM_EPS   1e-8f
// log2(0.9), log2(0.999) for beta^step = exp2(step * log2(beta))
#define L2_B1 (-0.15200309344504997f)
#define L2_B2 (-0.00144342518430764f)

#ifndef __has_builtin
#define __has_builtin(x) 0
#endif

#define GAS __attribute__((address_space(1)))
#define LAS __attribute__((address_space(3)))

typedef int v2i32 __attribute__((vector_size(2 * sizeof(int))));
typedef int v4i32 __attribute__((vector_size(4 * sizeof(int))));

#if __has_builtin(__builtin_amdgcn_global_load_async_to_lds_b128)
#define ASYNC_MODE 2     // native b128
#elif __has_builtin(__builtin_amdgcn_global_load_async_to_lds_b64)
#define ASYNC_MODE 1     // 2x b64 (known-good from round 1/2)
#else
#define ASYNC_MODE 0     // synchronous fallback
#endif

// 16 bytes per lane, global -> LDS, tracked by ASYNCcnt.
__device__ __forceinline__ void async_g2l_b128(const float* g, float* l) {
#if ASYNC_MODE == 2
    __builtin_amdgcn_global_load_async_to_lds_b128(
        (GAS v4i32*)(uintptr_t)g,   // global src (flat addr == AS1 addr)
        (LAS v4i32*)l,              // LDS dst (addrspacecast generic->AS3)
        0, 0);
#elif ASYNC_MODE == 1
    __builtin_amdgcn_global_load_async_to_lds_b64(
        (GAS v2i32*)(uintptr_t)g,     (LAS v2i32*)l,       0, 0);
    __builtin_amdgcn_global_load_async_to_lds_b64(
        (GAS v2i32*)(uintptr_t)(g+2), (LAS v2i32*)(l + 2), 0, 0);
#else
    *(float4*)l = *(const float4*)g;
#endif
}

__device__ __forceinline__ void wait_async_done() {
#if ASYNC_MODE != 0
#if __has_builtin(__builtin_amdgcn_s_wait_asynccnt)
    __builtin_amdgcn_s_wait_asynccnt(0);
#else
    asm volatile("s_wait_asynccnt 0" ::: "memory");
#endif
    asm volatile("" ::: "memory");   // compiler barrier: LDS now valid
#endif
}

// ---------------------------------------------------------------------------
// Phase 1: zero n floats (float4 main path + scalar tail).
// ---------------------------------------------------------------------------
__global__ void zero_floats(float* __restrict__ p, long long n) {
    long long i = (long long)blockIdx.x * blockDim.x + threadIdx.x;
    long long stride = (long long)gridDim.x * blockDim.x;
    long long n4 = n >> 2;
    float4* p4 = (float4*)p;
    float4 z = make_float4(0.f, 0.f, 0.f, 0.f);
    for (long long j = i; j < n4; j += stride) p4[j] = z;
    long long tail = n & 3;
    if (i < tail) p[(n4 << 2) + i] = 0.f;
}

// ---------------------------------------------------------------------------
// Phase 2: scatter.  16 lanes per gradient row; lane l owns elements
// 4l..4l+3 (one b128 load, four global_atomic_add_f32).
// ---------------------------------------------------------------------------
__global__ void __launch_bounds__(256)
scatter_grad(const int* __restrict__ idx,
             const float* __restrict__ grad,
             float* __restrict__ gsum,   // [V, 64] accumulator (= emb_new region)
             float* __restrict__ cnt,    // [V]     accumulator (= step_new region)
             int n_rows) {
    int t = blockIdx.x * blockDim.x + threadIdx.x;
    int row  = t >> 4;
    int lane = t & 15;
    if (row >= n_rows) return;

    int v = __ldg(idx + row);
    const float4 g = *(const float4*)(grad + (size_t)row * D_DIM + lane * 4);
    float* dst = gsum + (size_t)v * D_DIM + lane * 4;
    unsafeAtomicAdd(dst,     g.x);
    unsafeAtomicAdd(dst + 1, g.y);
    unsafeAtomicAdd(dst + 2, g.z);
    unsafeAtomicAdd(dst + 3, g.w);
    if (lane == 0) unsafeAtomicAdd(cnt + v, 1.0f);
}

// ---------------------------------------------------------------------------
// Phase 3: finalize.  16 lanes per table row (a wave covers 2 rows).
// gsum/cnt are read from the output regions they were accumulated into, then
// overwritten in place.  Streamed arrays staged through LDS with async
// global->LDS b128 loads; outputs stored as b128.
// ---------------------------------------------------------------------------
__global__ void __launch_bounds__(256)
adam_finalize(const float* __restrict__ emb,
              const float* __restrict__ step_in,
              const float* __restrict__ mem_in,
              const float* __restrict__ pow_in,
              float* __restrict__ out_emb,    // in: gsum,  out: emb_new
              float* __restrict__ out_step,   // in: cnt,   out: step_new
              float* __restrict__ out_mem,
              float* __restrict__ out_pow,
              int v_rows) {
    __shared__ float lbuf[16][4][D_DIM];      // 16 row-slots x 4 arrays x 256B = 16 KB

    int t = blockIdx.x * blockDim.x + threadIdx.x;
    int row  = t >> 4;
    int lane = t & 15;
    if (row >= v_rows) return;
    int slot = threadIdx.x >> 4;              // row slot within the block

    size_t off = (size_t)row * D_DIM + lane * 4;

    // Per-row scalars.  Read before the in-place stores below; each row is
    // owned by exactly one wave, and a wave issues (and waits on) all its
    // loads before any of its stores, so the in-place update is safe.
    float c = out_step[row];     // dedup count
    float s = step_in[row];

    // Stage the four streamed float4's per lane through LDS (ASYNCcnt path).
    async_g2l_b128(out_emb + off, &lbuf[slot][0][lane * 4]);   // gsum
    async_g2l_b128(emb     + off, &lbuf[slot][1][lane * 4]);
    async_g2l_b128(mem_in  + off, &lbuf[slot][2][lane * 4]);
    async_g2l_b128(pow_in  + off, &lbuf[slot][3][lane * 4]);
    wait_async_done();

    float4 gs = *(float4*)&lbuf[slot][0][lane * 4];
    float4 e  = *(float4*)&lbuf[slot][1][lane * 4];
    float4 m  = *(float4*)&lbuf[slot][2][lane * 4];
    float4 p  = *(float4*)&lbuf[slot][3][lane * 4];

    bool touched = (c > 0.0f);
    float s_new = touched ? (s + 1.0f) : s;

    float4 re, rm, rp;
    if (touched) {
        float inv = 1.0f / c;                 // mean gradient over duplicates
        float d1 = 1.0f - exp2f(s_new * L2_B1);   // 1 - b1^step
        float d2 = 1.0f - exp2f(s_new * L2_B2);   // 1 - b2^step
        float lr_d1 = ADAM_LR / d1;
        float rd2   = 1.0f / d2;

        float gv[4] = { gs.x * inv, gs.y * inv, gs.z * inv, gs.w * inv };
        float mi[4] = { m.x, m.y, m.z, m.w };
        float pi[4] = { p.x, p.y, p.z, p.w };
        float ei[4] = { e.x, e.y, e.z, e.w };
        float mo[4], po[4], eo[4];
#pragma unroll
        for (int k = 0; k < 4; ++k) {
            mo[k] = ADAM_B1 * mi[k] + (1.0f - ADAM_B1) * gv[k];
            po[k] = ADAM_B2 * pi[k] + (1.0f - ADAM_B2) * gv[k] * gv[k];
            eo[k] = ei[k] - lr_d1 * mo[k] / (sqrtf(po[k] * rd2) + ADAM_EPS);
        }
        rm = make_float4(mo[0], mo[1], mo[2], mo[3]);
        rp = make_float4(po[0], po[1], po[2], po[3]);
        re = make_float4(eo[0], eo[1], eo[2], eo[3]);
    } else {
        re = e; rm = m; rp = p;
    }

    *(float4*)(out_emb + off) = re;
    *(float4*)(out_mem + off) = rm;
    *(float4*)(out_pow + off) = rp;
    if (lane == 0) out_step[row] = s_new;
}

// ---------------------------------------------------------------------------
// Host launcher (graph-capture safe: kernels only, all on `stream`).
// ---------------------------------------------------------------------------
extern "C" void kernel_launch(void* const* d_in, const int* in_sizes, int n_in,
                              void* d_out, int out_size, void* d_ws, size_t ws_size,
                              hipStream_t stream) {
    const int*   idx   = (const int*)  d_in[0];
    const float* grad  = (const float*)d_in[1];
    const float* emb   = (const float*)d_in[2];
    const float* step  = (const float*)d_in[3];
    const float* mem   = (const float*)d_in[4];
    const float* pw    = (const float*)d_in[5];

    const int N = in_sizes[0];          // 524288 gradient rows
    const int V = in_sizes[3];          // 1e6 table rows

    float* out      = (float*)d_out;
    float* out_emb  = out;                              // [V,64] (gsum scratch)
    float* out_step = out + (size_t)V * D_DIM;          // [V]    (cnt scratch)
    float* out_mem  = out_step + V;                     // [V,64]
    float* out_pow  = out_mem + (size_t)V * D_DIM;      // [V,64]

    // Phase 1: zero accumulator regions (emb_new + step_new = V*65 floats).
    long long zero_n = (long long)V * (D_DIM + 1);
    zero_floats<<<8192, 256, 0, stream>>>(out_emb, zero_n);

    // Phase 2: scatter (16 threads per gradient row).
    long long sc_threads = (long long)N * 16;
    int sc_blocks = (int)((sc_threads + 255) / 256);
    scatter_grad<<<sc_blocks, 256, 0, stream>>>(idx, grad, out_emb, out_step, N);

    // Phase 3: finalize (16 threads per table row).
    long long fn_threads = (long long)V * 16;
    int fn_blocks = (int)((fn_threads + 255) / 256);
    adam_finalize<<<fn_blocks, 256, 0, stream>>>(emb, step, mem, pw,
                                                 out_emb, out_step, out_mem, out_pow, V);
}